// GraphAttentionNetwork_18708877541516
// MI455X (gfx1250) — compile-verified
//
#include <hip/hip_runtime.h>
#include <hip/hip_bf16.h>

#define N_HEADS   4
#define OUT_C_    64
#define HC        256       // N_HEADS * OUT_C
#define NEG_SLOPE 0.2f

typedef __attribute__((ext_vector_type(2))) float v2f;
typedef __attribute__((ext_vector_type(8))) float v8f;

// ---------- order-preserving uint encoding for float atomic max ----------
__device__ __forceinline__ unsigned f2ord(float f) {
  unsigned u = __float_as_uint(f);
  return (u & 0x80000000u) ? ~u : (u | 0x80000000u);
}
__device__ __forceinline__ float ord2f(unsigned u) {
  unsigned v = (u & 0x80000000u) ? (u & 0x7fffffffu) : ~u;
  return __uint_as_float(v);
}

// ---------- init: zero output accumulator, smax (ordered), denom ----------
__global__ void init_ws_kernel(float* __restrict__ out, unsigned* __restrict__ smax,
                               float* __restrict__ denom, int nOut, int nNH) {
  int i = blockIdx.x * blockDim.x + threadIdx.x;
  if (i < nOut) out[i] = 0.0f;
  if (i < nNH) { smax[i] = 0u; denom[i] = 0.0f; }
}

// ---------- xl = x @ W  via V_WMMA_F32_16X16X4_F32 ----------
// x: [N, 64] row-major, W: [64, 256] row-major, xl: [N, 256]
// One wave per 16x16 output tile; K loop of 16 WMMA issues (K=4 each).
__global__ void gemm_xw_wmma(const float* __restrict__ x, const float* __restrict__ W,
                             float* __restrict__ xl, int N) {
  const int lane  = threadIdx.x & 31;
  const int wave  = (blockIdx.x * blockDim.x + threadIdx.x) >> 5;
  const int colTiles = HC / 16;              // 16
  const int rowTile  = wave / colTiles;
  const int colTile  = wave % colTiles;
  const int row0 = rowTile * 16;
  const int col0 = colTile * 16;
  if (row0 >= N) return;                     // wave-uniform exit: EXEC all-ones inside

  const int mn    = lane & 15;               // M row (A) / N col (B,C)
  const int kpair = (lane >> 4) * 2;         // lanes 0-15: K=0,1 ; lanes 16-31: K=2,3

  v8f acc = {};
  #pragma unroll
  for (int k0 = 0; k0 < OUT_C_; k0 += 4) {
    v2f a, b;
    // A 16x4 f32: lane holds (M=mn, K=k0+kpair) and (M=mn, K=k0+kpair+1)
    a.x = x[(size_t)(row0 + mn) * OUT_C_ + k0 + kpair];
    a.y = x[(size_t)(row0 + mn) * OUT_C_ + k0 + kpair + 1];
    // B 4x16 f32: lane holds (K=k0+kpair, N=mn) and (K=k0+kpair+1, N=mn)
    b.x = W[(size_t)(k0 + kpair) * HC + col0 + mn];
    b.y = W[(size_t)(k0 + kpair + 1) * HC + col0 + mn];
    acc = __builtin_amdgcn_wmma_f32_16x16x4_f32(
        /*neg_a=*/false, a, /*neg_b=*/false, b,
        /*c_mod=*/(short)0, acc, /*reuse_a=*/false, /*reuse_b=*/false);
  }

  // C/D layout: VGPR v -> M = v (lanes 0-15) or v+8 (lanes 16-31); N = lane&15
  const int rbase = (lane >> 4) * 8;
  #pragma unroll
  for (int v = 0; v < 8; ++v) {
    xl[(size_t)(row0 + rbase + v) * HC + col0 + mn] = acc[v];
  }
}

// ---------- per-edge score + segment max (one wave32 per edge) ----------
__global__ void edge_scores_kernel(const float* __restrict__ xl,
                                   const long long* __restrict__ ei,
                                   const float* __restrict__ att,
                                   float* __restrict__ score,
                                   unsigned* __restrict__ smax_ord,
                                   int E) {
  const int wave = (blockIdx.x * blockDim.x + threadIdx.x) >> 5;
  const int lane = threadIdx.x & 31;
  if (wave >= E) return;
  const long long s = ei[wave];
  const long long d = ei[(size_t)E + wave];
  const float* __restrict__ xs = xl + (size_t)s * HC;
  const float* __restrict__ xd = xl + (size_t)d * HC;
  const int c = lane * 2;                       // 32 lanes x 2 ch = 64 channels
  #pragma unroll
  for (int h = 0; h < N_HEADS; ++h) {
    const int base = h * OUT_C_ + c;
    float f0 = xs[base]     + xd[base];
    float f1 = xs[base + 1] + xd[base + 1];
    f0 = (f0 > 0.0f) ? f0 : NEG_SLOPE * f0;     // LeakyReLU
    f1 = (f1 > 0.0f) ? f1 : NEG_SLOPE * f1;
    float p = f0 * att[base] + f1 * att[base + 1];
    #pragma unroll
    for (int off = 16; off > 0; off >>= 1)
      p += __shfl_down(p, off, 32);             // wave32 reduction
    if (lane == 0) {
      score[(size_t)wave * N_HEADS + h] = p;
      atomicMax(&smax_ord[(size_t)d * N_HEADS + h], f2ord(p));
    }
  }
}

// ---------- decode segment max; non-finite / untouched -> 0 ----------
__global__ void smax_finalize_kernel(unsigned* __restrict__ smax, int n) {
  int i = blockIdx.x * blockDim.x + threadIdx.x;
  if (i >= n) return;
  unsigned u = smax[i];
  float m = (u == 0u) ? 0.0f : ord2f(u);
  if (!isfinite(m)) m = 0.0f;
  ((float*)smax)[i] = m;                        // in-place reuse as float
}

// ---------- ex = exp(score - smax[dst]); denom[dst] += ex ----------
__global__ void edge_exp_kernel(const long long* __restrict__ ei,
                                const float* __restrict__ smax,
                                float* __restrict__ score,
                                float* __restrict__ denom,
                                int E) {
  int i = blockIdx.x * blockDim.x + threadIdx.x;
  if (i >= E * N_HEADS) return;
  const int e = i >> 2;                         // / N_HEADS (==4)
  const int h = i & (N_HEADS - 1);
  const long long d = ei[(size_t)E + e];
  float ex = __expf(score[i] - smax[(size_t)d * N_HEADS + h]);
  score[i] = ex;                                // overwrite score with ex
  atomicAdd(&denom[(size_t)d * N_HEADS + h], ex);
}

// ---------- out[dst] += alpha * xl[src] (one wave32 per edge) ----------
__global__ void edge_scatter_kernel(const float* __restrict__ xl,
                                    const long long* __restrict__ ei,
                                    const float* __restrict__ score,
                                    const float* __restrict__ denom,
                                    float* __restrict__ out,
                                    int E) {
  const int wave = (blockIdx.x * blockDim.x + threadIdx.x) >> 5;
  const int lane = threadIdx.x & 31;
  if (wave >= E) return;
  const long long s = ei[wave];
  const long long d = ei[(size_t)E + wave];
  const float* __restrict__ xs = xl + (size_t)s * HC;
  float* __restrict__ od = out + (size_t)d * HC;
  const int c = lane * 2;
  #pragma unroll
  for (int h = 0; h < N_HEADS; ++h) {
    float dn = denom[(size_t)d * N_HEADS + h];
    dn = (dn > 0.0f) ? dn : 1.0f;
    const float alpha = score[(size_t)wave * N_HEADS + h] / dn;
    const int base = h * OUT_C_ + c;
    atomicAdd(&od[base],     alpha * xs[base]);
    atomicAdd(&od[base + 1], alpha * xs[base + 1]);
  }
}

// ---------- out = relu(out + bias) ----------
__global__ void bias_relu_kernel(float* __restrict__ out, const float* __restrict__ bias, int n) {
  int i = blockIdx.x * blockDim.x + threadIdx.x;
  if (i >= n) return;
  float v = out[i] + bias[i & (HC - 1)];
  out[i] = (v > 0.0f) ? v : 0.0f;
}

extern "C" void kernel_launch(void* const* d_in, const int* in_sizes, int n_in,
                              void* d_out, int out_size, void* d_ws, size_t ws_size,
                              hipStream_t stream) {
  const float*     x    = (const float*)d_in[0];
  const long long* ei   = (const long long*)d_in[1];   // int64 edge_index [2, E]
  const float*     W    = (const float*)d_in[2];
  const float*     att  = (const float*)d_in[3];
  const float*     bias = (const float*)d_in[4];
  float*           out  = (float*)d_out;

  const int N = in_sizes[0] / OUT_C_;    // 50000
  const int E = in_sizes[1] / 2;         // 800000

  // workspace layout (~66 MB)
  float*    xl    = (float*)d_ws;                                  // N*256 f32
  float*    score = xl + (size_t)N * HC;                           // E*4   f32
  unsigned* smax  = (unsigned*)(score + (size_t)E * N_HEADS);      // N*4   u32->f32
  float*    denom = (float*)(smax + (size_t)N * N_HEADS);          // N*4   f32

  const int nOut = N * HC;
  const int nNH  = N * N_HEADS;

  init_ws_kernel<<<(nOut + 255) / 256, 256, 0, stream>>>(out, smax, denom, nOut, nNH);

  // one wave per 16x16 tile; 8 waves per 256-thread block (wave32)
  const int tiles = ((N + 15) / 16) * (HC / 16);
  gemm_xw_wmma<<<(tiles + 7) / 8, 256, 0, stream>>>(x, W, xl, N);

  const int edgeBlocks = (E + 7) / 8;    // 8 wave32 per block
  edge_scores_kernel<<<edgeBlocks, 256, 0, stream>>>(xl, ei, att, score, smax, E);
  smax_finalize_kernel<<<(nNH + 255) / 256, 256, 0, stream>>>(smax, nNH);
  edge_exp_kernel<<<(E * N_HEADS + 255) / 256, 256, 0, stream>>>(ei, (const float*)smax, score, denom, E);
  edge_scatter_kernel<<<edgeBlocks, 256, 0, stream>>>(xl, ei, score, denom, out, E);
  bias_relu_kernel<<<(nOut + 255) / 256, 256, 0, stream>>>(out, bias, nOut);
}